// CCGNN_90589450207918
// MI455X (gfx1250) — compile-verified
//
#include <hip/hip_runtime.h>
#include <math.h>

// CDNA5 / gfx1250: wave32, WMMA f32 16x16x4 (matches fp32 reference numerics;
// workload is bandwidth/atomic bound so no precision downgrade is needed).
// This revision adds Tensor Data Mover (TDM) staging of the weight tiles:
// tensor_load_to_lds + s_wait_tensorcnt, guarded so compilation still
// succeeds if the builtin is unavailable.

typedef float v2f __attribute__((ext_vector_type(2)));
typedef float v8f __attribute__((ext_vector_type(8)));
typedef float f4v __attribute__((ext_vector_type(4)));
typedef unsigned int u32x4 __attribute__((ext_vector_type(4)));
typedef int i32x4 __attribute__((ext_vector_type(4)));
typedef int i32x8 __attribute__((ext_vector_type(8)));

#define DD 128
#define TKC 64

#if defined(__gfx1250__) && __has_builtin(__builtin_amdgcn_tensor_load_to_lds)
#define USE_TDM 1
#else
#define USE_TDM 0
#endif

#if USE_TDM
// Build a 2D-tile D# (ISA 8.3/8.4) and issue TENSOR_LOAD_TO_LDS.
// tile = tile_rows x tile_ex f32 elements, global row stride = stride_e elems.
// All descriptor dwords forced uniform via readfirstlane (D# lives in SGPRs).
__device__ __forceinline__ void tdm_load_2d(unsigned lds_off, const void* gptr,
                                            unsigned tile_ex, unsigned tile_rows,
                                            unsigned stride_e)
{
  unsigned long long ga = (unsigned long long)gptr;
  unsigned long long st = (unsigned long long)stride_e;
  unsigned td0 = tile_ex, td1 = tile_rows;
  auto U = [](unsigned v) {
    return (unsigned)__builtin_amdgcn_readfirstlane((int)v);
  };
  u32x4 g0;
  g0[0] = U(1u);                                    // count=1 (valid user D#)
  g0[1] = U(lds_off);                               // LDS byte address
  g0[2] = U((unsigned)(ga & 0xFFFFFFFFu));          // global_addr[31:0]
  g0[3] = U((unsigned)((ga >> 32) & 0x01FFFFFFu) | (2u << 30)); // [56:32]|type=2
  i32x8 g1;
  g1[0] = (int)U(2u << 16);                         // data_size=4B, mask=0
  g1[1] = (int)U((td0 & 0xFFFFu) << 16);            // tensor_dim0[15:0]
  g1[2] = (int)U((td0 >> 16) | ((td1 & 0xFFFFu) << 16)); // td0 hi | td1 lo
  g1[3] = (int)U((td1 >> 16) | ((tile_ex & 0xFFFFu) << 16)); // td1 hi|tile_dim0
  g1[4] = (int)U(tile_rows & 0xFFFFu);              // tile_dim1 (tile_dim2=0)
  g1[5] = (int)U((unsigned)(st & 0xFFFFFFFFull));   // dim0_stride[31:0]
  g1[6] = (int)U((unsigned)((st >> 32) & 0xFFFFull)); // dim0_stride[47:32]
  g1[7] = 0;                                        // dim1_stride unused (2D)
  i32x4 z4 = {0, 0, 0, 0};
#if __clang_major__ >= 23
  i32x8 z8 = {0, 0, 0, 0, 0, 0, 0, 0};
  __builtin_amdgcn_tensor_load_to_lds(g0, g1, z4, z4, z8, 0);
#else
  __builtin_amdgcn_tensor_load_to_lds(g0, g1, z4, z4, 0);
#endif
}
#endif

// Cooperative fallback / non-TDM load of a TKC x 128 W chunk into LDS.
__device__ __forceinline__ void load_w_chunk(float* Ws, const float* W, int kc,
                                             int tid)
{
#pragma unroll
  for (int i = 0; i < 8; ++i) {
    int f  = tid + i * 256;
    int r  = f >> 5;
    int c4 = f & 31;
    *(f4v*)&Ws[r * DD + c4 * 4] = *(const f4v*)(W + (size_t)(kc + r) * DD + c4 * 4);
  }
}

// ---------------------------------------------------------------------------
// Generic GEMM: out[n x 128] = X[n x K] @ W[K x 128] (+ bias). n % 16 == 0,
// K % 64 == 0. Block = 256 threads = 8 waves; block owns a 16-row strip,
// wave w owns columns [16w, 16w+16). A tile + W chunk staged in LDS; W chunk
// staged by the TDM when available.
// WMMA f32 16x16x4 lane layout (ISA 7.12.2):
//   A (16x4): lane m = l&15, kb = 2*(l>>4); VGPR0=K=kb, VGPR1=K=kb+1
//   B (4x16): lane n = l&15, same kb split
//   C (16x16): VGPR r -> row r (lanes 0-15) / row r+8 (lanes 16-31), col l&15
// ---------------------------------------------------------------------------
__global__ __launch_bounds__(256) void k_gemm(
    const float* __restrict__ X, const float* __restrict__ W,
    const float* __restrict__ bias, float* __restrict__ out, int K)
{
  __shared__ __align__(16) float As[16 * TKC];
  __shared__ __align__(16) float Ws[TKC * DD];
  const int tid  = threadIdx.x;
  const int wave = tid >> 5;
  const int lane = tid & 31;
  const int row0 = blockIdx.x * 16;

  v8f c = {0.f,0.f,0.f,0.f,0.f,0.f,0.f,0.f};
  const int m    = lane & 15;
  const int kb   = (lane >> 4) << 1;
  const int ncol = wave * 16 + (lane & 15);

  for (int kc = 0; kc < K; kc += TKC) {
#if USE_TDM
    if (wave == 0) { // DMA the 64x128 W chunk; overlaps the A-tile loads below
      tdm_load_2d((unsigned)(unsigned long long)&Ws[0], W + (size_t)kc * DD,
                  DD, TKC, DD);
    }
#endif
    { // A tile: 16 x 64 floats = 256 float4, one per thread
      int r  = tid >> 4;          // 0..15
      int c4 = tid & 15;          // 16 float4 per row
      const float* src = X + (size_t)(row0 + r) * K + kc + c4 * 4;
      *(f4v*)&As[r * TKC + c4 * 4] = *(const f4v*)src;
      if (kc + TKC < K) __builtin_prefetch(src + TKC, 0, 1); // global_prefetch_b8
    }
#if USE_TDM
    if (wave == 0) __builtin_amdgcn_s_wait_tensorcnt(0);
#else
    load_w_chunk(Ws, W, kc, tid);
#endif
    __syncthreads();
    for (int k = 0; k < TKC; k += 4) {
      v2f a, b;
      a.x = As[m * TKC + k + kb];
      a.y = As[m * TKC + k + kb + 1];
      b.x = Ws[(k + kb) * DD + ncol];
      b.y = Ws[(k + kb + 1) * DD + ncol];
      c = __builtin_amdgcn_wmma_f32_16x16x4_f32(false, a, false, b, (short)0, c,
                                                false, false);
    }
    __syncthreads();
  }
  const int rbase = (lane >> 4) << 3;
  const float bv = bias ? bias[ncol] : 0.f;
  #pragma unroll
  for (int r = 0; r < 8; ++r)
    out[(size_t)(row0 + rbase + r) * DD + ncol] = c[r] + bv;
}

// ---------------------------------------------------------------------------
// Head GEMM with fused gather: h[s,:] = z(s0)@W1[0:128,:] + z(s1)@W1[128:256,:]
// + b1, where z(i) = (i < nA) ? zA[i] : zB[i-nA]. Avoids materializing uv.
// ---------------------------------------------------------------------------
__global__ __launch_bounds__(256) void k_head_gemm(
    const float* __restrict__ zA, const float* __restrict__ zB, int nA,
    const int* __restrict__ samples, const float* __restrict__ W1,
    const float* __restrict__ b1, float* __restrict__ h)
{
  __shared__ __align__(16) float As[16 * 256];
  __shared__ __align__(16) float Ws[TKC * DD];
  const int tid  = threadIdx.x;
  const int wave = tid >> 5;
  const int lane = tid & 31;
  const int s0   = blockIdx.x * 16;

  #pragma unroll
  for (int i = 0; i < 4; ++i) { // gather 16 x 256 floats = 1024 float4
    int f    = tid + i * 256;
    int r    = f >> 6;          // 0..15
    int c4   = f & 63;
    int half = c4 >> 5;         // 0 = u row, 1 = v row
    int cc   = (c4 & 31) << 2;
    int idx  = samples[2 * (s0 + r) + half];
    const float* base = (idx < nA) ? (zA + (size_t)idx * DD)
                                   : (zB + (size_t)(idx - nA) * DD);
    *(f4v*)&As[r * 256 + c4 * 4] = *(const f4v*)(base + cc);
  }

  v8f c = {0.f,0.f,0.f,0.f,0.f,0.f,0.f,0.f};
  const int m    = lane & 15;
  const int kb   = (lane >> 4) << 1;
  const int ncol = wave * 16 + (lane & 15);

  for (int kc = 0; kc < 256; kc += TKC) {
#if USE_TDM
    if (wave == 0) {
      tdm_load_2d((unsigned)(unsigned long long)&Ws[0], W1 + (size_t)kc * DD,
                  DD, TKC, DD);
      __builtin_amdgcn_s_wait_tensorcnt(0);
    }
#else
    load_w_chunk(Ws, W1, kc, tid);
#endif
    __syncthreads();
    for (int k = 0; k < TKC; k += 4) {
      v2f a, b;
      a.x = As[m * 256 + kc + k + kb];
      a.y = As[m * 256 + kc + k + kb + 1];
      b.x = Ws[(k + kb) * DD + ncol];
      b.y = Ws[(k + kb + 1) * DD + ncol];
      c = __builtin_amdgcn_wmma_f32_16x16x4_f32(false, a, false, b, (short)0, c,
                                                false, false);
    }
    __syncthreads();
  }
  const int rbase = (lane >> 4) << 3;
  const float bv = b1[ncol];
  #pragma unroll
  for (int r = 0; r < 8; ++r)
    h[(size_t)(s0 + rbase + r) * DD + ncol] = c[r] + bv;
}

// ------------------------------ GCN helpers --------------------------------
__global__ void k_fill(float* p, int n, float v) {
  int i = blockIdx.x * 256 + threadIdx.x;
  if (i < n) p[i] = v;
}

__global__ void k_deg_acc(float* deg, const int* __restrict__ dst,
                          const float* __restrict__ ew, int E) {
  int e = blockIdx.x * 256 + threadIdx.x;
  if (e < E) atomicAdd(&deg[dst[e]], ew ? ew[e] : 1.f);
}

__global__ void k_rsqrt_ip(float* deg, int n) {
  int i = blockIdx.x * 256 + threadIdx.x;
  if (i < n) deg[i] = rsqrtf(deg[i]);
}

// agg = dinv^2 * xw + b   (self term + bias, before edge scatter)
__global__ void k_agg_init(float* __restrict__ agg, const float* __restrict__ xw,
                           const float* __restrict__ dinv,
                           const float* __restrict__ b, int total) {
  int i = blockIdx.x * 256 + threadIdx.x;
  if (i >= total) return;
  int row = i >> 7, col = i & 127;
  float dv = dinv[row];
  agg[i] = dv * dv * xw[i] + b[col];
}

// one wave (32 lanes) per edge; each lane handles 4 columns via float4 load
__global__ __launch_bounds__(256) void k_edge_scatter(
    float* __restrict__ agg, const float* __restrict__ xw,
    const float* __restrict__ dinv, const int* __restrict__ src,
    const int* __restrict__ dst, const float* __restrict__ ew, int E)
{
  int wid  = (blockIdx.x * 256 + threadIdx.x) >> 5;
  int lane = threadIdx.x & 31;
  if (wid >= E) return;
  int s = src[wid], d = dst[wid];
  float w = ew ? ew[wid] : 1.f;
  float norm = dinv[s] * w * dinv[d];
  f4v v = *(const f4v*)(xw + (size_t)s * DD + lane * 4);
  float* ad = agg + (size_t)d * DD + lane * 4;
  atomicAdd(ad + 0, norm * v[0]);
  atomicAdd(ad + 1, norm * v[1]);
  atomicAdd(ad + 2, norm * v[2]);
  atomicAdd(ad + 3, norm * v[3]);
}

__global__ void k_relu(float* __restrict__ out, const float* __restrict__ in, int n) {
  int i = blockIdx.x * 256 + threadIdx.x;
  if (i < n) out[i] = fmaxf(in[i], 0.f);
}

__global__ void k_relu_split(float* __restrict__ a, float* __restrict__ b,
                             const float* __restrict__ in, int n, int nA) {
  int i = blockIdx.x * 256 + threadIdx.x;
  if (i >= (n << 7)) return;
  float v = fmaxf(in[i], 0.f);
  int row = i >> 7;
  if (row < nA) a[i] = v;
  else          b[i - (nA << 7)] = v;
}

// ------------------------------ Attention ----------------------------------
__global__ __launch_bounds__(256) void k_attn(
    const float* __restrict__ a, const float* __restrict__ b,
    const float* __restrict__ W1, const float* __restrict__ b1v,
    const float* __restrict__ w2, float* __restrict__ out, int n)
{
  __shared__ float sW[DD * 16];
  __shared__ float sb[16], sw[16];
  int tid = threadIdx.x;
  for (int i = tid; i < DD * 16; i += 256) sW[i] = W1[i];
  if (tid < 16) { sb[tid] = b1v[tid]; sw[tid] = w2[tid]; }
  __syncthreads();
  int i = blockIdx.x * 256 + tid;
  if (i >= n) return;
  const float* ar = a + (size_t)i * DD;
  const float* br = b + (size_t)i * DD;
  float s0 = 0.f, s1 = 0.f;
  for (int hh = 0; hh < 16; ++hh) {
    float a0 = sb[hh], a1 = sb[hh];
    for (int d = 0; d < DD; ++d) {
      float w = sW[d * 16 + hh];
      a0 += ar[d] * w;
      a1 += br[d] * w;
    }
    s0 += tanhf(a0) * sw[hh];
    s1 += tanhf(a1) * sw[hh];
  }
  float mx = fmaxf(s0, s1);
  float e0 = expf(s0 - mx), e1 = expf(s1 - mx);
  float inv = 1.f / (e0 + e1);
  float be0 = e0 * inv, be1 = e1 * inv;
  float* o = out + (size_t)i * DD;
  for (int d = 0; d < DD; ++d) o[d] = be0 * ar[d] + be1 * br[d];
}

// ------------------------------ BN / head epilogue -------------------------
__global__ __launch_bounds__(128) void k_colsum(
    const float* __restrict__ h, float* __restrict__ sum,
    float* __restrict__ sq, int NS)
{
  int c  = threadIdx.x;
  int r0 = blockIdx.x * 512;
  int re = min(r0 + 512, NS);
  float s = 0.f, ss = 0.f;
  for (int r = r0; r < re; ++r) {
    float v = h[(size_t)r * DD + c];
    s += v; ss += v * v;
  }
  atomicAdd(&sum[c], s);
  atomicAdd(&sq[c], ss);
}

// sum/sq -> scale/shift in place: hnorm = h*scale + shift
__global__ void k_bn_final(float* sum, float* sq, const float* __restrict__ g,
                           const float* __restrict__ bt, int NS)
{
  int c = threadIdx.x; // 128 threads
  float mu  = sum[c] / (float)NS;
  float var = sq[c] / (float)NS - mu * mu;
  float sc  = rsqrtf(var + 1e-5f) * g[c];
  sum[c] = sc;
  sq[c]  = bt[c] - mu * sc;
}

// wave per sample: normalize+relu, dot with W2 (shfl reduce), sigmoid, BCE sum
__global__ __launch_bounds__(256) void k_head_finish(
    const float* __restrict__ h, const float* __restrict__ scale,
    const float* __restrict__ shift, const float* __restrict__ W2,
    const float* __restrict__ b2, const float* __restrict__ labels,
    float* __restrict__ out_p, float* __restrict__ loss_acc, int NS)
{
  __shared__ float blk;
  if (threadIdx.x == 0) blk = 0.f;
  __syncthreads();
  int lane = threadIdx.x & 31;
  int s = blockIdx.x * 8 + (threadIdx.x >> 5);
  if (s < NS) {
    f4v hv = *(const f4v*)(h + (size_t)s * DD + lane * 4);
    int c = lane * 4;
    float acc = 0.f;
    #pragma unroll
    for (int j = 0; j < 4; ++j) {
      float hn = fmaxf(hv[j] * scale[c + j] + shift[c + j], 0.f);
      acc += hn * W2[c + j];
    }
    #pragma unroll
    for (int off = 16; off; off >>= 1) acc += __shfl_xor(acc, off, 32);
    if (lane == 0) {
      float p = 1.f / (1.f + expf(-(acc + b2[0])));
      if (out_p) out_p[s] = p;
      float pc = fminf(fmaxf(p, 1e-7f), 1.f - 1e-7f);
      float y = labels[s];
      atomicAdd(&blk, -(y * logf(pc) + (1.f - y) * log1pf(-pc)));
    }
  }
  __syncthreads();
  if (threadIdx.x == 0) atomicAdd(loss_acc, blk);
}

__global__ void k_total(const float* loss, float* out_total, float invNS) {
  out_total[0] = (loss[0] + 0.5f * (loss[1] + loss[2])) * invNS;
}

// ===========================================================================
extern "C" void kernel_launch(void* const* d_in, const int* in_sizes, int n_in,
                              void* d_out, int out_size, void* d_ws, size_t ws_size,
                              hipStream_t stream)
{
  (void)n_in; (void)out_size; (void)ws_size;
  constexpr int NRn = 20000, NPn = 20000, Fn = 512, Dn = 128, En = 640000,
                NSn = 200000;

  const float* xr      = (const float*)d_in[0];
  const float* xp      = (const float*)d_in[1];
  const int*   rr_src  = (const int*)d_in[2];
  const int*   rr_dst  = (const int*)d_in[3];
  const float* rr_attr = (const float*)d_in[4];
  const int*   pp_src  = (const int*)d_in[5];
  const int*   pp_dst  = (const int*)d_in[6];
  const float* pp_attr = (const float*)d_in[7];
  const int*   rp_src  = (const int*)d_in[8];
  const int*   rp_dst  = (const int*)d_in[9];
  const int*   samples = (const int*)d_in[10];
  const float* labels  = (const float*)d_in[11];

  auto L = [&](int i) { return (const float*)d_in[i]; };

  const float *proj_rr_W, *proj_rr_b, *proj_rpxr_W, *proj_rpxr_b,
              *proj_rpxp_W, *proj_rpxp_b, *proj_pp_W, *proj_pp_b;
  const float* enc_rr[6]; const float* enc_rp[6]; const float* enc_pp[6];
  const float *rat_W1, *rat_b1, *rat_w2, *pat_W1, *pat_b1, *pat_w2;
  struct HP { const float *W1, *b1, *g, *bt, *W2, *b2; } mlp, imlp, cmlp;

  if (in_sizes[12] == 2 * Dn * Dn) {
    // sorted-key tree flatten: c_mlp.W1 (256x128) at leaf 12
    auto head = [&](int b, HP& h) {
      h.W1 = L(b); h.W2 = L(b+1); h.b1 = L(b+2);
      h.b2 = L(b+3); h.bt = L(b+4); h.g = L(b+5);
    };
    head(12, cmlp); head(18, imlp); head(24, mlp);
    pat_W1 = L(30); pat_b1 = L(31); pat_w2 = L(32);
    for (int i = 0; i < 6; ++i) enc_pp[i] = L(33 + i);
    proj_pp_W = L(39); proj_pp_b = L(40);
    rat_W1 = L(41); rat_b1 = L(42); rat_w2 = L(43);
    for (int i = 0; i < 6; ++i) enc_rp[i] = L(44 + i);
    proj_rpxp_W = L(50); proj_rpxp_b = L(51);
    proj_rpxr_W = L(52); proj_rpxr_b = L(53);
    for (int i = 0; i < 6; ++i) enc_rr[i] = L(54 + i);
    proj_rr_W = L(60); proj_rr_b = L(61);
  } else {
    // insertion-order flatten: rr_proj.W (512x128) at leaf 12
    proj_rr_W   = L(12); proj_rr_b   = L(13);
    proj_rpxr_W = L(14); proj_rpxr_b = L(15);
    proj_rpxp_W = L(16); proj_rpxp_b = L(17);
    proj_pp_W   = L(18); proj_pp_b   = L(19);
    for (int i = 0; i < 6; ++i) enc_rr[i] = L(20 + i);
    for (int i = 0; i < 6; ++i) enc_rp[i] = L(26 + i);
    for (int i = 0; i < 6; ++i) enc_pp[i] = L(32 + i);
    rat_W1 = L(38); rat_b1 = L(39); rat_w2 = L(40);
    pat_W1 = L(41); pat_b1 = L(42); pat_w2 = L(43);
    auto head = [&](int b, HP& h) {
      h.W1 = L(b); h.b1 = L(b+1); h.g = L(b+2);
      h.bt = L(b+3); h.W2 = L(b+4); h.b2 = L(b+5);
    };
    head(44, mlp); head(50, imlp); head(56, cmlp);
  }

  // ---- workspace layout (floats); head h-buffer overlays encoder buffers --
  float* ws    = (float*)d_ws;
  float* hrb   = ws;                               // NR*D
  float* hpb   = hrb + (size_t)NRn * Dn;           // NP*D
  float* deg   = hpb + (size_t)NPn * Dn;           // NR+NP
  float* bnsum = deg + (NRn + NPn);                // 128
  float* bnsq  = bnsum + Dn;                       // 128
  float* lossv = bnsq + Dn;                        // 3 (+pad)
  float* bufX  = lossv + 64;                       // (NR+NP)*D
  float* bufXW = bufX + (size_t)(NRn + NPn) * Dn;  // (NR+NP)*D
  float* bufAGG= bufXW + (size_t)(NRn + NPn) * Dn; // (NR+NP)*D
  float* hbuf  = bufX;                             // NS*D (reuses enc region)
  (void)bufAGG;

  // ---- output layout: (out, total, rr_hr, rp_hr, pp_hp, rp_hp) ----
  float* out_p     = (float*)d_out;
  float* out_total = out_p + NSn;
  float* o_rr_hr   = out_total + 1;
  float* o_rp_hr   = o_rr_hr + (size_t)NRn * Dn;
  float* o_pp_hp   = o_rp_hr + (size_t)NRn * Dn;
  float* o_rp_hp   = o_pp_hp + (size_t)NPn * Dn;

  auto cdiv = [](int a, int b) { return (a + b - 1) / b; };

  auto run_encoder = [&](float* x, const int* src, const int* dst,
                         const float* ew, int n, const float* const* wb,
                         float* outFull, float* outA, float* outB, int nA) {
    k_fill<<<cdiv(n, 256), 256, 0, stream>>>(deg, n, 1.f);
    k_deg_acc<<<cdiv(En, 256), 256, 0, stream>>>(deg, dst, ew, En);
    k_rsqrt_ip<<<cdiv(n, 256), 256, 0, stream>>>(deg, n);
    for (int l = 0; l < 3; ++l) {
      k_gemm<<<n / 16, 256, 0, stream>>>(x, wb[2 * l], nullptr, bufXW, Dn);
      k_agg_init<<<cdiv(n * Dn, 256), 256, 0, stream>>>(bufAGG, bufXW, deg,
                                                        wb[2 * l + 1], n * Dn);
      k_edge_scatter<<<En / 8, 256, 0, stream>>>(bufAGG, bufXW, deg, src, dst,
                                                 ew, En);
      if (l < 2)
        k_relu<<<cdiv(n * Dn, 256), 256, 0, stream>>>(x, bufAGG, n * Dn);
      else if (outFull)
        k_relu<<<cdiv(n * Dn, 256), 256, 0, stream>>>(outFull, bufAGG, n * Dn);
      else
        k_relu_split<<<cdiv(n * Dn, 256), 256, 0, stream>>>(outA, outB, bufAGG,
                                                            n, nA);
    }
  };

  auto run_head = [&](const float* zA, const float* zB, int nA, const HP& hp,
                      float* outp, float* lossp) {
    k_fill<<<1, 256, 0, stream>>>(bnsum, 2 * Dn, 0.f);
    k_fill<<<1, 32, 0, stream>>>(lossp, 1, 0.f);
    k_head_gemm<<<NSn / 16, 256, 0, stream>>>(zA, zB, nA, samples, hp.W1, hp.b1,
                                              hbuf);
    k_colsum<<<cdiv(NSn, 512), 128, 0, stream>>>(hbuf, bnsum, bnsq, NSn);
    k_bn_final<<<1, 128, 0, stream>>>(bnsum, bnsq, hp.g, hp.bt, NSn);
    k_head_finish<<<NSn / 8, 256, 0, stream>>>(hbuf, bnsum, bnsq, hp.W2, hp.b2,
                                               labels, outp, lossp, NSn);
  };

  // ---- rr branch ----
  k_gemm<<<NRn / 16, 256, 0, stream>>>(xr, proj_rr_W, proj_rr_b, bufX, Fn);
  run_encoder(bufX, rr_src, rr_dst, rr_attr, NRn, enc_rr, o_rr_hr, nullptr,
              nullptr, 0);
  // ---- pp branch ----
  k_gemm<<<NPn / 16, 256, 0, stream>>>(xp, proj_pp_W, proj_pp_b, bufX, Fn);
  run_encoder(bufX, pp_src, pp_dst, pp_attr, NPn, enc_pp, o_pp_hp, nullptr,
              nullptr, 0);
  // ---- rp branch (concat [rp_xr; rp_xp], ones edge weights) ----
  k_gemm<<<NRn / 16, 256, 0, stream>>>(xr, proj_rpxr_W, proj_rpxr_b, bufX, Fn);
  k_gemm<<<NPn / 16, 256, 0, stream>>>(xp, proj_rpxp_W, proj_rpxp_b,
                                       bufX + (size_t)NRn * Dn, Fn);
  run_encoder(bufX, rp_src, rp_dst, nullptr, NRn + NPn, enc_rp, nullptr,
              o_rp_hr, o_rp_hp, NRn);

  // ---- attention ----
  k_attn<<<cdiv(NRn, 256), 256, 0, stream>>>(o_rr_hr, o_rp_hr, rat_W1, rat_b1,
                                             rat_w2, hrb, NRn);
  k_attn<<<cdiv(NPn, 256), 256, 0, stream>>>(o_pp_hp, o_rp_hp, pat_W1, pat_b1,
                                             pat_w2, hpb, NPn);

  // ---- heads: z = [hr;hp], rp_h = [rp_hr;rp_hp], z_c = [rr_hr;pp_hp] ----
  run_head(hrb, hpb, NRn, mlp, out_p, lossv + 0);
  run_head(o_rp_hr, o_rp_hp, NRn, imlp, nullptr, lossv + 1);
  run_head(o_rr_hr, o_pp_hp, NRn, cmlp, nullptr, lossv + 2);

  k_total<<<1, 1, 0, stream>>>(lossv, out_total, 1.f / (float)NSn);
}